// SoftReordering_67869073212262
// MI455X (gfx1250) — compile-verified
//
#include <hip/hip_runtime.h>
#include <math.h>

typedef __attribute__((ext_vector_type(2))) float v2f;
typedef __attribute__((ext_vector_type(8))) float v8f;
typedef __attribute__((ext_vector_type(4))) int   v4i;

#define Bsz 16
#define Ssz 1024
#define Esz 1024
#define Wsz 7
#define Ksz (Wsz * Esz)   // 7168
#define HW  3
#define WROW 1032         // LDS row stride in floats (8-dword pad -> conflict-free ds_load_b64)

#if defined(__AMDGCN__) && __has_builtin(__builtin_amdgcn_global_load_async_to_lds_b128)
#define USE_ASYNC_LDS 1
#else
#define USE_ASYNC_LDS 0
#endif

typedef __attribute__((address_space(1))) v4i gv4i;   // global
typedef __attribute__((address_space(3))) v4i lv4i;   // LDS

__device__ __forceinline__ void wait_async_copies() {
#if USE_ASYNC_LDS
#if __has_builtin(__builtin_amdgcn_s_wait_asynccnt)
    __builtin_amdgcn_s_wait_asynccnt(0);
#else
    asm volatile("s_wait_asynccnt 0x0" ::: "memory");
#endif
#endif
}

// Stage weight[t][v][w*1024 .. w*1024+1023] (7 rows x 4 KB) into LDS, coalesced b128.
__device__ __forceinline__ void stage_chunk(const float* __restrict__ wtile, int w,
                                            float* dstbuf, int tid)
{
    #pragma unroll
    for (int i = 0; i < Wsz; ++i) {
        const int idx = i * 256 + tid;          // 1792 x 16B transfers
        const int v   = idx >> 8;
        const int seg = (idx & 255) << 2;       // float offset within the 1024-row
        const float* g = wtile + (size_t)v * Ksz + (size_t)w * Esz + seg;
        float* l = dstbuf + v * WROW + seg;
#if USE_ASYNC_LDS
        __builtin_amdgcn_global_load_async_to_lds_b128((gv4i*)g, (lv4i*)l, 0, 0);
#else
        *(float4*)l = *(const float4*)g;
#endif
    }
}

// One workgroup per position t. 8 wave32 waves split the K=7168 reduction.
// WMMA f32 16x16x4: M = batch(16), N = gate v (7 of 16 columns used), K = (w,e).
__global__ __launch_bounds__(256)
void soft_reorder_kernel(const float* __restrict__ x,
                         const float* __restrict__ weight,
                         const float* __restrict__ bias,
                         float* __restrict__ out)
{
    const int t    = blockIdx.x;
    const int tid  = threadIdx.x;
    const int wave = tid >> 5;          // 0..7
    const int lane = tid & 31;
    const int laneLo  = lane & 15;      // M row (batch) for A, N col (gate v) for B
    const int hi      = lane >> 4;      // 0: K=0,1  1: K=2,3
    const int laneOff = hi * 2;
    const int vc      = laneLo < Wsz ? laneLo : (Wsz - 1);  // clamp: cols 7..15 = garbage, never read

    __shared__ float wbuf[2][Wsz * WROW];   // double-buffered weight chunk (2 x 28.2 KB)
    __shared__ float red[256];              // 16x16 cross-wave logit reduction
    __shared__ float gates[16][8];

    red[tid] = 0.0f;

    const float* wtile = weight + (size_t)t * Wsz * Ksz;
    const float* xbase = x + (size_t)laneLo * Ssz * Esz + laneOff;
    const int e0base = wave * (Esz / 8);    // each wave owns a 128-wide e slice

    v8f acc0 = {};
    v8f acc1 = {};

    stage_chunk(wtile, 0, wbuf[0], tid);    // prologue: chunk w=0

    for (int w = 0; w < Wsz; ++w) {
        const int cur = w & 1;
        wait_async_copies();                // my copies for chunk w are done
        __syncthreads();                    // => everyone's copies done; prev compute done
        if (w + 1 < Wsz)
            stage_chunk(wtile, w + 1, wbuf[cur ^ 1], tid);  // overlap next chunk

        const int row = t + w - HW;
        if (row >= 0 && row < Ssz) {        // zero-padded tap contributes nothing
            const float* xr = xbase + (size_t)row * Esz + e0base;
            const float* wl = &wbuf[cur][vc * WROW + e0base + laneOff];
            __builtin_prefetch(xr + Esz, 0, 0);   // next tap's x slice -> global_prefetch
            #pragma unroll
            for (int e = 0; e < Esz / 8; e += 8) {
                v2f a0 = *(const v2f*)(xr + e);
                v2f a1 = *(const v2f*)(xr + e + 4);
                v2f b0 = *(const v2f*)(wl + e);
                v2f b1 = *(const v2f*)(wl + e + 4);
                acc0 = __builtin_amdgcn_wmma_f32_16x16x4_f32(false, a0, false, b0,
                                                             (short)0, acc0, false, false);
                acc1 = __builtin_amdgcn_wmma_f32_16x16x4_f32(false, a1, false, b1,
                                                             (short)0, acc1, false, false);
            }
        }
    }

    // D layout: element (m,n) at accumulator VGPR r with m = r + 8*hi, n = laneLo.
    #pragma unroll
    for (int r = 0; r < 8; ++r) {
        const float partial = acc0[r] + acc1[r];
        atomicAdd(&red[(r + hi * 8) * 16 + laneLo], partial);   // ds_add_f32
    }
    __syncthreads();

    // logits -> sigmoid gates (16 batches x 7 taps)
    if (tid < Bsz * Wsz) {
        const int b = tid / Wsz;
        const int v = tid % Wsz;
        const float logit = red[b * 16 + v] + bias[t * Wsz + v];
        gates[b][v] = 1.0f / (1.0f + __expf(-logit));
    }
    __syncthreads();

    // Phase 2: out[b,t,e] = tanh( sum_w gates[b,w] * x[b, t+w-3, e] ); coalesced, L2 hits.
    for (int idx = tid; idx < Bsz * Esz; idx += 256) {
        const int b = idx >> 10;
        const int e = idx & (Esz - 1);
        float s = 0.0f;
        #pragma unroll
        for (int w = 0; w < Wsz; ++w) {
            const int row = t + w - HW;
            if (row >= 0 && row < Ssz) {
                s += gates[b][w] * x[((size_t)b * Ssz + row) * Esz + e];
            }
        }
        out[((size_t)b * Ssz + t) * Esz + e] = tanhf(s);
    }
}

extern "C" void kernel_launch(void* const* d_in, const int* in_sizes, int n_in,
                              void* d_out, int out_size, void* d_ws, size_t ws_size,
                              hipStream_t stream) {
    const float* x      = (const float*)d_in[0];   // (16, 1024, 1024) f32
    const float* weight = (const float*)d_in[1];   // (1024, 7, 7168) f32
    const float* bias   = (const float*)d_in[2];   // (1024, 7) f32
    float* out = (float*)d_out;                    // (16, 1024, 1024) f32

    dim3 grid(Ssz);    // one workgroup per t -> weight streamed from HBM exactly once
    dim3 block(256);   // 8 wave32 waves
    hipLaunchKernelGGL(soft_reorder_kernel, grid, block, 0, stream, x, weight, bias, out);
}